// GaussianRasterizer_40054865002576
// MI455X (gfx1250) — compile-verified
//
#include <hip/hip_runtime.h>
#include <hip/hip_bf16.h>
#include <math.h>

// ---------------------------------------------------------------------------
// Gaussian splat forward for MI455X (gfx1250, wave32).
//
// Cost model: 256*256*512 ~= 33.5M pixel-gaussian pairs, ~12 VALU + 1 exp
// each => compute bound, ~us-scale; memory traffic (<1 MB) irrelevant at
// 23.3 TB/s. The per-pixel cumprod is sequential, so the blend loop stays
// VALU. WMMA is used where it is exact and natural: a single
// V_WMMA_F32_16X16X4_F32 per 16 gaussians computes BOTH 4x4 projective
// transforms (view & proj stacked as rows 0..7 of the A matrix), landing
// each gaussian's p_view/p_hom in its owning lane's D registers.
// ---------------------------------------------------------------------------

typedef __attribute__((ext_vector_type(2))) float v2f;
typedef __attribute__((ext_vector_type(8))) float v8f;

#define GN 512
#define GH 256
#define GW 256
#define FXC 256.0f      // W / (2*tanx)
#define FYC 256.0f      // H / (2*tany)
#define LIMX 0.65f      // 1.3 * tanx
#define LIMY 0.65f

// Workspace (floats):
//   raw SoA   [10][GN] at 0      : tz, conA, conB, conC, pixx, pixy, op, r, g, b
//   sorted SoA [9][GN] at 10*GN  : conA, conB, conC, pixx, pixy, op, r, g, b

__global__ void gs_preprocess(const float* __restrict__ means3D,
                              const float* __restrict__ opac,
                              const float* __restrict__ cols,
                              const float* __restrict__ scales,
                              const float* __restrict__ rots,
                              const float* __restrict__ viewm,
                              const float* __restrict__ projm,
                              float* __restrict__ ws,
                              int* __restrict__ radii_out) {
    const int lane = threadIdx.x;        // one wave per block
    const int row  = lane & 15;
    const int kb   = (lane < 16) ? 0 : 2;

    // A (16x4): rows 0..3 = viewmatrix, rows 4..7 = projmatrix, rows 8..15 = 0.
    // A-layout: lanes 0-15 hold K=0,1 ; lanes 16-31 hold K=2,3 (VGPR0/VGPR1).
    float a0 = 0.0f, a1 = 0.0f;
    if (row < 4)      { a0 = viewm[row * 4 + kb];       a1 = viewm[row * 4 + kb + 1]; }
    else if (row < 8) { a0 = projm[(row - 4) * 4 + kb]; a1 = projm[(row - 4) * 4 + kb + 1]; }

    // B (4x16): column g = hom coords (x,y,z,1) of gaussian g.
    // B-layout: lanes 0-15 hold rows K=0,1 ; lanes 16-31 hold rows K=2,3.
    const int g = blockIdx.x * 16 + row;
    const float mx = means3D[g * 3 + 0];
    const float my = means3D[g * 3 + 1];
    const float mz = means3D[g * 3 + 2];
    float b0, b1;
    if (lane < 16) { b0 = mx; b1 = my; }
    else           { b0 = mz; b1 = 1.0f; }

    v2f A; A.x = a0; A.y = a1;
    v2f B; B.x = b0; B.y = b1;
    v8f C = {};
    // D[i][g]: rows 0..3 = p_view components, rows 4..7 = p_hom components.
    // Lane g (0..15) owns column g => D vgpr r = row r for its gaussian.
    v8f D = __builtin_amdgcn_wmma_f32_16x16x4_f32(
        /*neg_a=*/false, A, /*neg_b=*/false, B,
        /*c_mod=*/(short)0, C, /*reuse_a=*/false, /*reuse_b=*/false);

    if (lane >= 16) return;   // upper half held zero rows (8..15)

    const float pvx = D[0], pvy = D[1], tz = D[2];
    const float ph0 = D[4], ph1 = D[5], ph3 = D[7];

    // --- quaternion -> rotation ---
    const float q0 = rots[g * 4 + 0], q1 = rots[g * 4 + 1];
    const float q2 = rots[g * 4 + 2], q3 = rots[g * 4 + 3];
    const float qn = rsqrtf(q0 * q0 + q1 * q1 + q2 * q2 + q3 * q3);
    const float rq = q0 * qn, x = q1 * qn, y = q2 * qn, z = q3 * qn;
    const float R00 = 1.0f - 2.0f * (y * y + z * z), R01 = 2.0f * (x * y - rq * z), R02 = 2.0f * (x * z + rq * y);
    const float R10 = 2.0f * (x * y + rq * z), R11 = 1.0f - 2.0f * (x * x + z * z), R12 = 2.0f * (y * z - rq * x);
    const float R20 = 2.0f * (x * z - rq * y), R21 = 2.0f * (y * z + rq * x), R22 = 1.0f - 2.0f * (x * x + y * y);

    float s0 = scales[g * 3 + 0]; s0 *= s0;
    float s1 = scales[g * 3 + 1]; s1 *= s1;
    float s2 = scales[g * 3 + 2]; s2 *= s2;

    // Sigma = R diag(s^2) R^T (symmetric)
    const float S00 = R00 * R00 * s0 + R01 * R01 * s1 + R02 * R02 * s2;
    const float S01 = R00 * R10 * s0 + R01 * R11 * s1 + R02 * R12 * s2;
    const float S02 = R00 * R20 * s0 + R01 * R21 * s1 + R02 * R22 * s2;
    const float S11 = R10 * R10 * s0 + R11 * R11 * s1 + R12 * R12 * s2;
    const float S12 = R10 * R20 * s0 + R11 * R21 * s1 + R12 * R22 * s2;
    const float S22 = R20 * R20 * s0 + R21 * R21 * s1 + R22 * R22 * s2;

    // --- EWA Jacobian ---
    const float itz = 1.0f / tz;
    const float txc = fminf(LIMX, fmaxf(-LIMX, pvx * itz)) * tz;
    const float tyc = fminf(LIMY, fmaxf(-LIMY, pvy * itz)) * tz;
    const float J00 = FXC * itz, J02 = -FXC * txc * itz * itz;
    const float J11 = FYC * itz, J12 = -FYC * tyc * itz * itz;

    // Tm = J @ view[:3,:3]
    const float T00 = J00 * viewm[0] + J02 * viewm[8];
    const float T01 = J00 * viewm[1] + J02 * viewm[9];
    const float T02 = J00 * viewm[2] + J02 * viewm[10];
    const float T10 = J11 * viewm[4] + J12 * viewm[8];
    const float T11 = J11 * viewm[5] + J12 * viewm[9];
    const float T12 = J11 * viewm[6] + J12 * viewm[10];

    // cov2d = Tm Sigma Tm^T
    const float u00 = T00 * S00 + T01 * S01 + T02 * S02;
    const float u01 = T00 * S01 + T01 * S11 + T02 * S12;
    const float u02 = T00 * S02 + T01 * S12 + T02 * S22;
    const float u10 = T10 * S00 + T11 * S01 + T12 * S02;
    const float u11 = T10 * S01 + T11 * S11 + T12 * S12;
    const float u12 = T10 * S02 + T11 * S12 + T12 * S22;
    const float ca = u00 * T00 + u01 * T01 + u02 * T02 + 0.3f;
    const float cb = u00 * T10 + u01 * T11 + u02 * T12;
    const float cc = u10 * T10 + u11 * T11 + u12 * T12 + 0.3f;

    const float det = ca * cc - cb * cb;
    const bool valid = (tz > 0.2f) && (det > 0.0f);
    const float inv_det = 1.0f / det;
    const float conA = cc * inv_det, conB = -cb * inv_det, conC = ca * inv_det;

    const float pwv = 1.0f / (ph3 + 1e-7f);
    const float pix_x = ((ph0 * pwv + 1.0f) * (float)GW - 1.0f) * 0.5f;
    const float pix_y = ((ph1 * pwv + 1.0f) * (float)GH - 1.0f) * 0.5f;

    const float mid = 0.5f * (ca + cc);
    const float lam = mid + sqrtf(fmaxf(0.1f, mid * mid - det));
    radii_out[g] = valid ? (int)ceilf(3.0f * sqrtf(lam)) : 0;

    ws[0 * GN + g] = tz;
    ws[1 * GN + g] = valid ? conA : 0.0f;
    ws[2 * GN + g] = valid ? conB : 0.0f;
    ws[3 * GN + g] = valid ? conC : 0.0f;
    ws[4 * GN + g] = valid ? pix_x : 0.0f;
    ws[5 * GN + g] = valid ? pix_y : 0.0f;
    ws[6 * GN + g] = valid ? opac[g] : 0.0f;   // invalid => alpha 0, matches mask
    ws[7 * GN + g] = cols[g * 3 + 0];
    ws[8 * GN + g] = cols[g * 3 + 1];
    ws[9 * GN + g] = cols[g * 3 + 2];
}

// Stable O(N^2) rank sort by tz, scatter params into sorted SoA.
__global__ void gs_sort(float* __restrict__ ws) {
    __shared__ float stz[GN];
    const int i = threadIdx.x;
    const float tz = ws[i];
    stz[i] = tz;
    __syncthreads();
    int rank = 0;
    for (int j = 0; j < GN; ++j) {
        const float o = stz[j];
        rank += (o < tz) || (o == tz && j < i);   // stable (matches argsort)
    }
    float* __restrict__ dst = ws + 10 * GN;
    #pragma unroll
    for (int a = 0; a < 9; ++a)
        dst[a * GN + rank] = ws[(a + 1) * GN + i];
}

// One thread per pixel; sorted gaussians staged through LDS in chunks of 64.
__global__ void gs_composite(const float* __restrict__ ws,
                             const float* __restrict__ bg,
                             float* __restrict__ out) {
    const int p = blockIdx.x * blockDim.x + threadIdx.x;   // 65536 pixels exactly
    const float px = (float)(p & (GW - 1));
    const float py = (float)(p >> 8);                      // W == 256
    const float* __restrict__ S = ws + 10 * GN;

    __shared__ float sm[9][64];
    float T = 1.0f, cr = 0.0f, cg = 0.0f, cbv = 0.0f;

    for (int base = 0; base < GN; base += 64) {
        __syncthreads();
        for (int i = threadIdx.x; i < 9 * 64; i += blockDim.x) {
            const int arr = i >> 6, idx = i & 63;
            sm[arr][idx] = S[arr * GN + base + idx];
        }
        __syncthreads();
        #pragma unroll 4
        for (int k = 0; k < 64; ++k) {
            const float cA = sm[0][k], cB = sm[1][k], cC = sm[2][k];
            const float dx = sm[3][k] - px;
            const float dy = sm[4][k] - py;
            const float power = -0.5f * (cA * dx * dx + cC * dy * dy) - cB * dx * dy;
            float alpha = fminf(0.99f, sm[5][k] * __expf(power));
            if (power > 0.0f || alpha < (1.0f / 255.0f)) alpha = 0.0f;
            const float Tn = T * (1.0f - alpha);
            const float w = (Tn >= 1.0e-4f) ? (alpha * T) : 0.0f;
            cr  += w * sm[6][k];
            cg  += w * sm[7][k];
            cbv += w * sm[8][k];
            T = Tn;
        }
    }
    out[0 * GH * GW + p] = cr  + T * bg[0];
    out[1 * GH * GW + p] = cg  + T * bg[1];
    out[2 * GH * GW + p] = cbv + T * bg[2];
}

extern "C" void kernel_launch(void* const* d_in, const int* in_sizes, int n_in,
                              void* d_out, int out_size, void* d_ws, size_t ws_size,
                              hipStream_t stream) {
    const float* means3D = (const float*)d_in[0];
    // d_in[1] = means2D (unused by the reference math)
    const float* opac    = (const float*)d_in[2];
    const float* cols    = (const float*)d_in[3];
    const float* scales  = (const float*)d_in[4];
    const float* rots    = (const float*)d_in[5];
    const float* viewm   = (const float*)d_in[6];
    const float* projm   = (const float*)d_in[7];
    const float* bg      = (const float*)d_in[8];

    float* out   = (float*)d_out;
    int*   radii = (int*)d_out + 3 * GH * GW;   // int32 bits in the tail
    float* ws    = (float*)d_ws;

    gs_preprocess<<<GN / 16, 32, 0, stream>>>(means3D, opac, cols, scales, rots,
                                              viewm, projm, ws, radii);
    gs_sort<<<1, GN, 0, stream>>>(ws);
    gs_composite<<<(GH * GW) / 256, 256, 0, stream>>>(ws, bg, out);

    (void)in_sizes; (void)n_in; (void)out_size; (void)ws_size;
}